// GCN_multi_scale_5446018531914
// MI455X (gfx1250) — compile-verified
//
#include <hip/hip_runtime.h>
#include <hip/hip_bf16.h>

typedef __attribute__((ext_vector_type(2))) float v2f;
typedef __attribute__((ext_vector_type(8))) float v8f;

#define N_NODES 100000
#define N_EDGES 3200000
#define N_GRAPHS 64
#define D_IN 128
#define HID 128
#define D_STAT 32
#define N_CLASSES 10

// ---------------------------------------------------------------------------
// Utility: zero a float buffer (d_ws is poisoned; must re-zero every call)
// ---------------------------------------------------------------------------
__global__ void __launch_bounds__(256) k_zero(float* __restrict__ p, int n) {
    int i = blockIdx.x * 256 + threadIdx.x;
    if (i < n) p[i] = 0.0f;
}

// ---------------------------------------------------------------------------
// Pair-interleave a 128x128 weight matrix so a B fragment (W[k][c], W[k+1][c])
// becomes one aligned b64 load:  Wi[(k>>1)*256 + c*2 + (k&1)] = W[k*128 + c]
// ---------------------------------------------------------------------------
__global__ void __launch_bounds__(256) k_interleave(const float* __restrict__ W,
                                                    float* __restrict__ Wi) {
    int t = blockIdx.x * 256 + threadIdx.x;
    if (t < 128 * 128) {
        int k = t >> 7, c = t & 127;
        Wi[(k >> 1) * 256 + c * 2 + (k & 1)] = W[t];
    }
}

// ---------------------------------------------------------------------------
// Degree of dst (in-degree); dinv = rsqrt(deg + 1)  [self loop]
// ---------------------------------------------------------------------------
__global__ void __launch_bounds__(256) k_deg(const int* __restrict__ dst,
                                             float* __restrict__ deg, int nE) {
    int e = blockIdx.x * 256 + threadIdx.x;
    if (e < nE) atomicAdd(&deg[dst[e]], 1.0f);
}

__global__ void __launch_bounds__(256) k_dinv(const float* __restrict__ deg,
                                              float* __restrict__ dinv, int n) {
    int i = blockIdx.x * 256 + threadIdx.x;
    if (i < n) dinv[i] = rsqrtf(deg[i] + 1.0f);
}

__global__ void __launch_bounds__(256) k_batch_count(const int* __restrict__ batch,
                                                     float* __restrict__ cnt, int n) {
    int i = blockIdx.x * 256 + threadIdx.x;
    if (i < n) atomicAdd(&cnt[batch[i]], 1.0f);
}

// ---------------------------------------------------------------------------
// Fused: H = X @ W  (V_WMMA_F32_16X16X4_F32)  and  agg = H*dinv^2 + b.
// Block = 256 = 8 waves; wave owns one 16-row strip, loops all 8 col tiles
// per K-step (A fragment reused 8x).  B fragments come from the interleaved
// copy Wi as single b64 loads -> 9 VMEM issues per 8 WMMAs.
// Fragment layouts per CDNA5 ISA 7.12.2:
//   A (16x4 f32): lane L -> M = L%16, VGPR v -> K = k0 + v + 2*(L>=16)
//   B (4x16 f32): lane L -> N = L%16, VGPR v -> K = k0 + v + 2*(L>=16)
//   C/D         : lane L -> N = L%16, VGPR r -> M = r + 8*(L>=16)
// ---------------------------------------------------------------------------
__global__ void __launch_bounds__(256) k_gemm128_fused(const float* __restrict__ X,
                                                       const float* __restrict__ Wi,
                                                       const float* __restrict__ dinv,
                                                       const float* __restrict__ bias,
                                                       float* __restrict__ H,
                                                       float* __restrict__ agg,
                                                       int Mtiles) {
    const int wave = threadIdx.x >> 5;
    const int lane = threadIdx.x & 31;
    const int tile = blockIdx.x * 8 + wave;
    if (tile >= Mtiles) return;            // wave-uniform: EXEC all-ones inside

    const int lm   = lane & 15;
    const int koff = (lane >> 4) << 1;     // 0 (lanes 0-15) or 2 (lanes 16-31)
    const int row  = tile * 16 + lm;

    v8f acc[8];
#pragma unroll
    for (int n = 0; n < 8; ++n) acc[n] = (v8f){0, 0, 0, 0, 0, 0, 0, 0};

    const float*  __restrict__ xrow = X + (size_t)row * 128;
    const float2* __restrict__ wiv  = (const float2*)Wi;   // [64][128] float2

    for (int k0 = 0; k0 < 128; k0 += 4) {
        const float2 av = *(const float2*)(xrow + k0 + koff);   // b64
        v2f A; A.x = av.x; A.y = av.y;
        const float2* __restrict__ bp = wiv + (size_t)((k0 + koff) >> 1) * 128 + lm;
#pragma unroll
        for (int n = 0; n < 8; ++n) {
            const float2 bv = bp[n * 16];                       // b64
            v2f B; B.x = bv.x; B.y = bv.y;
            acc[n] = __builtin_amdgcn_wmma_f32_16x16x4_f32(
                false, A, false, B, (short)0, acc[n], false, false);
        }
    }

    // epilogue: write H and agg = H*dinv^2 + b (self-loop term, inits agg)
    const int rbase = tile * 16 + ((lane >> 4) << 3);  // +0 or +8
    float dv2[8];
#pragma unroll
    for (int r = 0; r < 8; ++r) {
        float d = dinv[rbase + r];
        dv2[r] = d * d;
    }
#pragma unroll
    for (int n = 0; n < 8; ++n) {
        const float bcol = bias[n * 16 + lm];
        float* __restrict__ hp = H   + (size_t)rbase * 128 + n * 16 + lm;
        float* __restrict__ ap = agg + (size_t)rbase * 128 + n * 16 + lm;
#pragma unroll
        for (int r = 0; r < 8; ++r) {
            float hv = acc[n][r];
            hp[(size_t)r * 128] = hv;
            ap[(size_t)r * 128] = hv * dv2[r] + bcol;
        }
    }
}

// ---------------------------------------------------------------------------
// For each edge: agg[dst,:] += h[src,:] * dinv[src]*dinv[dst]
// one wave per edge; coalesced b128 gather + 4 dword f32 atomics per lane.
// h and agg (51 MB each) live in L2 -> this is L2-atomic-throughput bound.
// ---------------------------------------------------------------------------
__global__ void __launch_bounds__(256) k_edge_scatter(const int* __restrict__ src,
                                                      const int* __restrict__ dst,
                                                      const float* __restrict__ dinv,
                                                      const float* __restrict__ h,
                                                      float* __restrict__ agg, int nE) {
    int lane = threadIdx.x & 31;
    int e = blockIdx.x * 8 + (threadIdx.x >> 5);
    if (e >= nE) return;
    int s = src[e], d = dst[e];
    float norm = dinv[s] * dinv[d];
    const float4 hv = *(const float4*)(h + (size_t)s * 128 + lane * 4);
    float* ap = agg + (size_t)d * 128 + lane * 4;
    atomicAdd(ap + 0, hv.x * norm);
    atomicAdd(ap + 1, hv.y * norm);
    atomicAdd(ap + 2, hv.z * norm);
    atomicAdd(ap + 3, hv.w * norm);
}

// ---------------------------------------------------------------------------
// psum[batch[i],:] += relu(h[i,:])   (mean done at classify time)
// ---------------------------------------------------------------------------
__global__ void __launch_bounds__(256) k_pool_acc(const float* __restrict__ h,
                                                  const int* __restrict__ batch,
                                                  float* __restrict__ psum, int n) {
    int t = blockIdx.x * 256 + threadIdx.x;
    int node = t >> 5;
    int c = (t & 31) << 2;
    if (node < n) {
        int g = batch[node];
        const float4 hv = *(const float4*)(h + (size_t)node * 128 + c);
        float* pp = psum + (size_t)g * 128 + c;
        atomicAdd(pp + 0, fmaxf(hv.x, 0.0f));
        atomicAdd(pp + 1, fmaxf(hv.y, 0.0f));
        atomicAdd(pp + 2, fmaxf(hv.z, 0.0f));
        atomicAdd(pp + 3, fmaxf(hv.w, 0.0f));
    }
}

// ---------------------------------------------------------------------------
// s[g,c] = relu(stat[g,:] @ Ws[:,c] + bs[c])   (64x32 @ 32x128 — tiny)
// ---------------------------------------------------------------------------
__global__ void __launch_bounds__(256) k_stat_mlp(const float* __restrict__ stat,
                                                  const float* __restrict__ Ws,
                                                  const float* __restrict__ bs,
                                                  float* __restrict__ s) {
    int t = blockIdx.x * 256 + threadIdx.x;
    if (t < N_GRAPHS * HID) {
        int g = t >> 7, c = t & 127;
        float acc = bs[c];
        for (int k = 0; k < D_STAT; ++k)
            acc += stat[g * D_STAT + k] * Ws[k * HID + c];
        s[t] = fmaxf(acc, 0.0f);
    }
}

// ---------------------------------------------------------------------------
// out[g,cls] = concat(p1/cnt, p2/cnt, p3/cnt, s)[g,:] @ Wc[:,cls] + bc[cls]
// ---------------------------------------------------------------------------
__global__ void __launch_bounds__(256) k_classify(const float* __restrict__ p1,
                                                  const float* __restrict__ p2,
                                                  const float* __restrict__ p3,
                                                  const float* __restrict__ s,
                                                  const float* __restrict__ cnt,
                                                  const float* __restrict__ Wc,
                                                  const float* __restrict__ bc,
                                                  float* __restrict__ out) {
    int t = blockIdx.x * 256 + threadIdx.x;
    if (t < N_GRAPHS * N_CLASSES) {
        int g = t / N_CLASSES, cls = t % N_CLASSES;
        float inv = 1.0f / fmaxf(cnt[g], 1.0f);
        float acc = bc[cls];
        for (int c = 0; c < HID; ++c)
            acc += p1[g * HID + c] * inv * Wc[c * N_CLASSES + cls];
        for (int c = 0; c < HID; ++c)
            acc += p2[g * HID + c] * inv * Wc[(HID + c) * N_CLASSES + cls];
        for (int c = 0; c < HID; ++c)
            acc += p3[g * HID + c] * inv * Wc[(2 * HID + c) * N_CLASSES + cls];
        for (int c = 0; c < HID; ++c)
            acc += s[g * HID + c] * Wc[(3 * HID + c) * N_CLASSES + cls];
        out[t] = acc;
    }
}

// ---------------------------------------------------------------------------
extern "C" void kernel_launch(void* const* d_in, const int* in_sizes, int n_in,
                              void* d_out, int out_size, void* d_ws, size_t ws_size,
                              hipStream_t stream) {
    const float* x    = (const float*)d_in[0];
    const int*   ei   = (const int*)d_in[1];
    const float* stat = (const float*)d_in[2];
    const int*   bat  = (const int*)d_in[3];
    const float* W1 = (const float*)d_in[4];  const float* b1 = (const float*)d_in[5];
    const float* W2 = (const float*)d_in[6];  const float* b2 = (const float*)d_in[7];
    const float* W3 = (const float*)d_in[8];  const float* b3 = (const float*)d_in[9];
    const float* Ws = (const float*)d_in[10]; const float* bs = (const float*)d_in[11];
    const float* Wc = (const float*)d_in[12]; const float* bc = (const float*)d_in[13];
    float* out = (float*)d_out;

    const int N = N_NODES, E = N_EDGES;
    const int* src = ei;          // edge_index[0]
    const int* dst = ei + E;      // edge_index[1]

    // workspace layout (floats)
    float* ws   = (float*)d_ws;
    size_t NF   = (size_t)N * HID;
    float* Htmp = ws;                         // [N,128]  X@W scratch
    float* bufA = Htmp + NF;                  // [N,128]  h1 / h3
    float* bufB = bufA + NF;                  // [N,128]  h2
    float* deg  = bufB + NF;                  // [N]
    float* dinv = deg + N;                    // [N]
    float* psum = dinv + N;                   // [3,64,128]
    float* cnt  = psum + 3 * N_GRAPHS * HID;  // [64]
    float* sbuf = cnt + N_GRAPHS;             // [64,128]
    float* Wi1  = sbuf + N_GRAPHS * HID;      // [64,128,2] interleaved weights
    float* Wi2  = Wi1 + 128 * 128;
    float* Wi3  = Wi2 + 128 * 128;

    const int Mtiles = N / 16;               // 6250 exactly
    dim3 gGemm((Mtiles + 7) / 8);
    dim3 gNode32(((size_t)N * 32 + 255) / 256);
    dim3 gEdgeW((E + 7) / 8);
    dim3 gW((128 * 128 + 255) / 256);
    int smallZ = 3 * N_GRAPHS * HID + N_GRAPHS;

    // --- graph-constant prep (deterministic each call) ---
    k_zero<<<(N + 255) / 256, 256, 0, stream>>>(deg, N);
    k_zero<<<(smallZ + 255) / 256, 256, 0, stream>>>(psum, smallZ);
    k_interleave<<<gW, 256, 0, stream>>>(W1, Wi1);
    k_interleave<<<gW, 256, 0, stream>>>(W2, Wi2);
    k_interleave<<<gW, 256, 0, stream>>>(W3, Wi3);
    k_deg<<<(E + 255) / 256, 256, 0, stream>>>(dst, deg, E);
    k_dinv<<<(N + 255) / 256, 256, 0, stream>>>(deg, dinv, N);
    k_batch_count<<<(N + 255) / 256, 256, 0, stream>>>(bat, cnt, N);

    // --- layer 1 ---
    k_gemm128_fused<<<gGemm, 256, 0, stream>>>(x, Wi1, dinv, b1, Htmp, bufA, Mtiles);
    k_edge_scatter<<<gEdgeW, 256, 0, stream>>>(src, dst, dinv, Htmp, bufA, E);
    k_pool_acc<<<gNode32, 256, 0, stream>>>(bufA, bat, psum, N);

    // --- layer 2 ---
    k_gemm128_fused<<<gGemm, 256, 0, stream>>>(bufA, Wi2, dinv, b2, Htmp, bufB, Mtiles);
    k_edge_scatter<<<gEdgeW, 256, 0, stream>>>(src, dst, dinv, Htmp, bufB, E);
    k_pool_acc<<<gNode32, 256, 0, stream>>>(bufB, bat, psum + N_GRAPHS * HID, N);

    // --- layer 3 (reuse bufA) ---
    k_gemm128_fused<<<gGemm, 256, 0, stream>>>(bufB, Wi3, dinv, b3, Htmp, bufA, Mtiles);
    k_edge_scatter<<<gEdgeW, 256, 0, stream>>>(src, dst, dinv, Htmp, bufA, E);
    k_pool_acc<<<gNode32, 256, 0, stream>>>(bufA, bat, psum + 2 * N_GRAPHS * HID, N);

    // --- head ---
    k_stat_mlp<<<(N_GRAPHS * HID + 255) / 256, 256, 0, stream>>>(stat, Ws, bs, sbuf);
    k_classify<<<(N_GRAPHS * N_CLASSES + 255) / 256, 256, 0, stream>>>(
        psum, psum + N_GRAPHS * HID, psum + 2 * N_GRAPHS * HID, sbuf, cnt, Wc, bc, out);
}